// Bigk_Rep_62912680952558
// MI455X (gfx1250) — compile-verified
//
#include <hip/hip_runtime.h>
#include <cstdint>

// Problem geometry (fixed by the reference)
#define HH 256
#define WW 256
#define CC 64

// Tiling
#define TILE_W 64
#define TILE_H 32
#define HALO_Y 3
#define HALO_XL 4                    // left x-halo padded to 4 so global b128 chunks are 16B aligned
#define IN_H   (TILE_H + 6)          // 38 rows
#define LDSP   72                    // LDS row pitch (floats): covers gx in [tx0-4, tx0+67]
#define CHUNKS_PER_ROW (LDSP / 4)    // 18 b128 chunks per row
#define NTHREADS 256

typedef float v4f __attribute__((ext_vector_type(4)));

// ---------------------------------------------------------------------------
// Fused reparameterized depthwise conv:
//   out = dwconv7(x, w7+pad(w5)+pad(w3)+pad(w1), b7+b5+b3+b1)
// CDNA5 path: b128 async global->LDS DMA (ASYNCcnt) for the haloed tile,
// non-temporal b128 stores for the streaming output.
// ---------------------------------------------------------------------------
__global__ __launch_bounds__(NTHREADS) void dwconv7_rep_fused(
    const float* __restrict__ x,
    const float* __restrict__ w7, const float* __restrict__ b7,
    const float* __restrict__ w1, const float* __restrict__ b1,
    const float* __restrict__ w3, const float* __restrict__ b3,
    const float* __restrict__ w5, const float* __restrict__ b5,
    float* __restrict__ out)
{
    __shared__ float tile[IN_H * LDSP];   // LDS col c  <->  gx = tx0 - 4 + c

    const int tid = threadIdx.x;
    const int tx0 = blockIdx.x * TILE_W;
    const int ty0 = blockIdx.y * TILE_H;
    const int zc  = blockIdx.z;              // b*CC + c
    const int c   = zc & (CC - 1);

    const float* xin = x + (size_t)zc * (HH * WW);

    // ---- Stage haloed tile into LDS via async b128 DMA; zero OOB chunks. ----
    // gx0 = tx0 - 4 + 4*cx is 0 mod 4, so each 4-col chunk is entirely
    // inside or entirely outside the image in x; rows likewise in y.
    // Each LDS chunk is written exactly once (async OR ds_store) -> one
    // wait + barrier suffices, no zero/overwrite race.
    for (int j = tid; j < IN_H * CHUNKS_PER_ROW; j += NTHREADS) {
        const int row = j / CHUNKS_PER_ROW;
        const int cx  = j - row * CHUNKS_PER_ROW;
        const int gy  = ty0 - HALO_Y + row;
        const int gx0 = tx0 - HALO_XL + (cx << 2);
        float* lp = &tile[row * LDSP + (cx << 2)];
        if ((unsigned)gy < (unsigned)HH && (unsigned)gx0 <= (unsigned)(WW - 4)) {
            const float* g = xin + gy * WW + gx0;          // 16B-aligned
            const unsigned lds_off = (unsigned)(uintptr_t)lp;  // low 32 bits = LDS offset
            asm volatile("global_load_async_to_lds_b128 %0, %1, off"
                         :: "v"(lds_off), "v"(g) : "memory");
        } else {
            *(v4f*)lp = (v4f){0.f, 0.f, 0.f, 0.f};         // zero padding
        }
    }

    // ---- While DMA is in flight: build merged 7x7 weights (uniform loads) ----
    float wk[49];
    #pragma unroll
    for (int ky = 0; ky < 7; ++ky) {
        #pragma unroll
        for (int kx = 0; kx < 7; ++kx) {
            float s = w7[c * 49 + ky * 7 + kx];
            if (ky >= 1 && ky <= 5 && kx >= 1 && kx <= 5)
                s += w5[c * 25 + (ky - 1) * 5 + (kx - 1)];
            if (ky >= 2 && ky <= 4 && kx >= 2 && kx <= 4)
                s += w3[c * 9 + (ky - 2) * 3 + (kx - 2)];
            if (ky == 3 && kx == 3)
                s += w1[c];
            wk[ky * 7 + kx] = s;
        }
    }
    const float bias = b7[c] + b1[c] + b3[c] + b5[c];

    asm volatile("s_wait_asynccnt 0" ::: "memory");
    __syncthreads();

    // ---- Register-blocked compute: each thread does 4(x) x 2(y) outputs ----
    const int lx = (tid & 15) << 2;   // 0,4,...,60
    const int ly = (tid >> 4) << 1;   // 0,2,...,30

    float acc0[4] = {bias, bias, bias, bias};
    float acc1[4] = {bias, bias, bias, bias};

    #pragma unroll
    for (int r = 0; r < 8; ++r) {
        // Window of 12 aligned floats; output x-offset lx+j, tap kx reads
        // LDS col (lx + kx + j) + (HALO_XL - 3) = lx + 1 + kx + j.
        const float* rp = &tile[(ly + r) * LDSP + lx];
        const v4f p0 = *(const v4f*)(rp);
        const v4f p1 = *(const v4f*)(rp + 4);
        const v4f p2 = *(const v4f*)(rp + 8);
        const float v[12] = {p0.x, p0.y, p0.z, p0.w,
                             p1.x, p1.y, p1.z, p1.w,
                             p2.x, p2.y, p2.z, p2.w};
        if (r < 7) {                       // output row ly, ky = r
            #pragma unroll
            for (int kx = 0; kx < 7; ++kx) {
                const float w = wk[r * 7 + kx];
                #pragma unroll
                for (int jj = 0; jj < 4; ++jj)
                    acc0[jj] = fmaf(w, v[1 + kx + jj], acc0[jj]);
            }
        }
        if (r >= 1) {                      // output row ly+1, ky = r-1
            #pragma unroll
            for (int kx = 0; kx < 7; ++kx) {
                const float w = wk[(r - 1) * 7 + kx];
                #pragma unroll
                for (int jj = 0; jj < 4; ++jj)
                    acc1[jj] = fmaf(w, v[1 + kx + jj], acc1[jj]);
            }
        }
    }

    // ---- Coalesced non-temporal b128 stores (output is write-once stream) ----
    const int ox = tx0 + lx;
    const int oy = ty0 + ly;
    float* o = out + (size_t)zc * (HH * WW) + (size_t)oy * WW + ox;
    __builtin_nontemporal_store((v4f){acc0[0], acc0[1], acc0[2], acc0[3]}, (v4f*)o);
    __builtin_nontemporal_store((v4f){acc1[0], acc1[1], acc1[2], acc1[3]}, (v4f*)(o + WW));
}

// ---------------------------------------------------------------------------
extern "C" void kernel_launch(void* const* d_in, const int* in_sizes, int n_in,
                              void* d_out, int out_size, void* d_ws, size_t ws_size,
                              hipStream_t stream) {
    (void)n_in; (void)d_ws; (void)ws_size; (void)out_size;
    const float* x  = (const float*)d_in[0];
    const float* w7 = (const float*)d_in[1];
    const float* b7 = (const float*)d_in[2];
    const float* w1 = (const float*)d_in[3];
    const float* b1 = (const float*)d_in[4];
    const float* w3 = (const float*)d_in[5];
    const float* b3 = (const float*)d_in[6];
    const float* w5 = (const float*)d_in[7];
    const float* b5 = (const float*)d_in[8];
    float* out = (float*)d_out;

    const int B = in_sizes[0] / (CC * HH * WW);   // 16
    dim3 grid(WW / TILE_W, HH / TILE_H, B * CC);  // 4 x 8 x 1024
    dwconv7_rep_fused<<<grid, NTHREADS, 0, stream>>>(
        x, w7, b7, w1, b1, w3, b3, w5, b5, out);
}